// FullyConnectedTP_29085518528898
// MI455X (gfx1250) — compile-verified
//
#include <hip/hip_runtime.h>

// Fully-connected e3nn tensor product, per-edge weights, on gfx1250.
// Design: HBM-bound on streaming 13824 f32 weights/edge (55KB), so:
//   stage 1 (cheap):  T_p[u,k] = pw * sum_ij C_p[ijk] u1[u,i] v[j]  -> LDS
//   stage 2 (heavy):  out[w,k] += sum_u W_p[u,w] T_p[u,k]  via V_WMMA_F32_16X16X4_F32
// One wave per edge; A operand = weights loaded directly from global in WMMA
// A-layout (each byte read exactly once); B operand = T from LDS; fp32 end-to-end.

typedef float v2f __attribute__((ext_vector_type(2)));
typedef float v8f __attribute__((ext_vector_type(8)));

#define WAVES_PER_BLOCK 4
#define TROWS 384           // T rows per wave, each padded to 8 floats

__device__ __forceinline__ v8f wmma4(v2f a, v2f b, v8f c) {
  // D(16x16,f32) = A(16x4,f32) * B(4x16,f32) + C
  return __builtin_amdgcn_wmma_f32_16x16x4_f32(false, a, false, b, (short)0, c,
                                               false, false);
}

// One path's contribution: acc[t] += W_p^T(16xK tile) * T_p(Kx16)
// A layout (f32 16x4): lanes 0-15 M=0..15 hold K=0,1; lanes 16-31 hold K=2,3
// B layout (f32 4x16): lanes 0-15 N=0..15 hold K=0,1; lanes 16-31 hold K=2,3
// B loads are unconditional (clamped in-bounds address); the zero for padded
// N-columns is applied to the loaded VALUE (v_cndmask), so no EXEC juggling.
template <int TILES, int MUL3, int NCOLS>
__device__ __forceinline__ void path_mm(const float* __restrict__ wp,
                                        const float* __restrict__ Tw,
                                        int rowbase, int K, int lane, v8f* acc) {
  const int kq = (lane >> 4) << 1;   // 0 or 2
  const int nn = lane & 15;
  const int nc = nn & 7;             // always-in-bounds column within padded row
  const bool live = (nn < NCOLS);
  for (int s = 0; s < K; s += 4) {
    const int ur = s + kq;
    const float tx = Tw[(rowbase + ur) * 8 + nc];
    const float ty = Tw[(rowbase + ur + 1) * 8 + nc];
    v2f b;
    b.x = live ? tx : 0.0f;
    b.y = live ? ty : 0.0f;
#pragma unroll
    for (int t = 0; t < TILES; ++t) {
      v2f a;
      a.x = wp[(ur)*MUL3 + t * 16 + nn];
      a.y = wp[(ur + 1) * MUL3 + t * 16 + nn];
      acc[t] = wmma4(a, b, acc[t]);
    }
  }
}

__global__ void __launch_bounds__(WAVES_PER_BLOCK * 32)
fctp_kernel(const float* __restrict__ u, const float* __restrict__ v,
            const float* __restrict__ wgt, float* __restrict__ out, int nEdges) {
  __shared__ float Tlds[WAVES_PER_BLOCK * TROWS * 8];
  const int wave = threadIdx.x >> 5;
  const int lane = threadIdx.x & 31;
  const int z = blockIdx.x * WAVES_PER_BLOCK + wave;
  float* Tw = &Tlds[wave * TROWS * 8];

  // path weights pw = sqrt((2l3+1)/fan): fan = 112 / 144 / 128
  const float pw0 = 0.09449111825230680f;   // sqrt(1/112)
  const float pw1 = 0.14433756729740645f;   // sqrt(3/144)
  const float pw2 = 0.19764235376052370f;   // sqrt(5/128)
  const float k1 = 0.08333333333333333f;    // pw1/sqrt(3)
  const float k2 = 0.08838834764831845f;    // pw2/sqrt(5)
  const float k4 = 0.05455447255899826f;    // pw0/sqrt(3)
  const float k9 = 0.04225771273642583f;    // pw0/sqrt(5)
  // wigner_3j(1,1,2) nonzeros (unit Frobenius), basis m=-l..l (y,z,x)/(xy,yz,z0,xz,x2-y2)
  const float c1 = 0.31622776601683794f;    // 1/sqrt(10)
  const float c2 = 0.36514837167011072f;    // 2/sqrt(30)
  const float c3 = 0.18257418583505536f;    // 1/sqrt(30)
  // wigner_3j(2,2,2) nonzeros (fully symmetric)
  const float A2 = 0.23904572186687872f;    // sqrt(70)/35
  const float B2 = 0.11952286093343936f;    // A2/2
  const float D2 = 0.20701966780270626f;    // sqrt(210)/70

  if (z < nEdges) {
    const float* uz = u + (size_t)z * 240;
    const float* vz = v + (size_t)z * 9;

    const float V0 = vz[0];
    float V1[3], V2[5];
    V1[0] = vz[1]; V1[1] = vz[2]; V1[2] = vz[3];
#pragma unroll
    for (int j = 0; j < 5; ++j) V2[j] = vz[4 + j];

    const float a0 = uz[lane];
    const float a0b = uz[32 + lane];
    float U1[3];
#pragma unroll
    for (int i = 0; i < 3; ++i) U1[i] = uz[64 + 3 * lane + i];

    // ---- T rows: P0[0] P1[64] P2[128] P3[192] P4[224] P5[256] P6[288]
    //              P7[320] P8[336] P9[352] P10[368]
    Tw[(lane)*8] = pw0 * a0 * V0;                       // P0 (0,0,0)
    Tw[(lane + 32) * 8] = pw0 * a0b * V0;
#pragma unroll
    for (int k = 0; k < 3; ++k) {                       // P1 (0,1,1)
      Tw[(64 + lane) * 8 + k] = k1 * a0 * V1[k];
      Tw[(96 + lane) * 8 + k] = k1 * a0b * V1[k];
    }
#pragma unroll
    for (int k = 0; k < 5; ++k) {                       // P2 (0,2,2)
      Tw[(128 + lane) * 8 + k] = k2 * a0 * V2[k];
      Tw[(160 + lane) * 8 + k] = k2 * a0b * V2[k];
    }
#pragma unroll
    for (int k = 0; k < 3; ++k)                         // P3 (1,0,1)
      Tw[(192 + lane) * 8 + k] = k1 * U1[k] * V0;
    Tw[(224 + lane) * 8] =                              // P4 (1,1,0)
        k4 * (U1[0] * V1[0] + U1[1] * V1[1] + U1[2] * V1[2]);
    // P5 (1,1,2)
    Tw[(256 + lane) * 8 + 0] = pw2 * (c1 * (U1[2] * V1[0] + U1[0] * V1[2]));
    Tw[(256 + lane) * 8 + 1] = pw2 * (c1 * (U1[0] * V1[1] + U1[1] * V1[0]));
    Tw[(256 + lane) * 8 + 2] =
        pw2 * (c2 * U1[1] * V1[1] - c3 * (U1[0] * V1[0] + U1[2] * V1[2]));
    Tw[(256 + lane) * 8 + 3] = pw2 * (c1 * (U1[2] * V1[1] + U1[1] * V1[2]));
    Tw[(256 + lane) * 8 + 4] = pw2 * (c1 * (U1[2] * V1[2] - U1[0] * V1[0]));
    // P6 (1,2,1): C121[i,j,k] = C112[i,k,j]
    Tw[(288 + lane) * 8 + 0] =
        pw1 * (c1 * (U1[2] * V2[0] + U1[1] * V2[1]) - c3 * U1[0] * V2[2] -
               c1 * U1[0] * V2[4]);
    Tw[(288 + lane) * 8 + 1] =
        pw1 * (c1 * U1[0] * V2[1] + c2 * U1[1] * V2[2] + c1 * U1[2] * V2[3]);
    Tw[(288 + lane) * 8 + 2] =
        pw1 * (c1 * U1[0] * V2[0] - c3 * U1[2] * V2[2] + c1 * U1[1] * V2[3] +
               c1 * U1[2] * V2[4]);

    if (lane < 16) {
      float U2[5];
#pragma unroll
      for (int i = 0; i < 5; ++i) U2[i] = uz[160 + 5 * lane + i];
#pragma unroll
      for (int k = 0; k < 5; ++k)                       // P7 (2,0,2)
        Tw[(320 + lane) * 8 + k] = k2 * U2[k] * V0;
      // P8 (2,1,1): C211[I,j,k] = C112[j,k,I]
      Tw[(336 + lane) * 8 + 0] =
          pw1 * (c1 * (V1[2] * U2[0] + V1[1] * U2[1]) - c3 * V1[0] * U2[2] -
                 c1 * V1[0] * U2[4]);
      Tw[(336 + lane) * 8 + 1] =
          pw1 * (c1 * V1[0] * U2[1] + c2 * V1[1] * U2[2] + c1 * V1[2] * U2[3]);
      Tw[(336 + lane) * 8 + 2] =
          pw1 * (c1 * V1[0] * U2[0] - c3 * V1[2] * U2[2] + c1 * V1[1] * U2[3] +
                 c1 * V1[2] * U2[4]);
      Tw[(352 + lane) * 8] =                            // P9 (2,2,0)
          k9 * (U2[0] * V2[0] + U2[1] * V2[1] + U2[2] * V2[2] +
                U2[3] * V2[3] + U2[4] * V2[4]);
      // P10 (2,2,2)
      Tw[(368 + lane) * 8 + 0] =
          pw2 * (-A2 * (U2[0] * V2[2] + U2[2] * V2[0]) +
                 D2 * (U2[1] * V2[3] + U2[3] * V2[1]));
      Tw[(368 + lane) * 8 + 1] =
          pw2 * (B2 * (U2[1] * V2[2] + U2[2] * V2[1]) +
                 D2 * (U2[0] * V2[3] + U2[3] * V2[0]) -
                 D2 * (U2[1] * V2[4] + U2[4] * V2[1]));
      Tw[(368 + lane) * 8 + 2] =
          pw2 * (A2 * U2[2] * V2[2] - A2 * U2[0] * V2[0] +
                 B2 * U2[1] * V2[1] + B2 * U2[3] * V2[3] -
                 A2 * U2[4] * V2[4]);
      Tw[(368 + lane) * 8 + 3] =
          pw2 * (B2 * (U2[2] * V2[3] + U2[3] * V2[2]) +
                 D2 * (U2[0] * V2[1] + U2[1] * V2[0]) +
                 D2 * (U2[3] * V2[4] + U2[4] * V2[3]));
      Tw[(368 + lane) * 8 + 4] =
          pw2 * (-A2 * (U2[2] * V2[4] + U2[4] * V2[2]) - D2 * U2[1] * V2[1] +
                 D2 * U2[3] * V2[3]);
    }
  }
  __syncthreads();
  if (z >= nEdges) return;

  const float* wz = wgt + (size_t)z * 13824;
  float* oz = out + (size_t)z * 240;
  const int nn = lane & 15;
  const int mh = (lane >> 4) << 3;  // 0 or 8
  const v8f vzero = {0.f, 0.f, 0.f, 0.f, 0.f, 0.f, 0.f, 0.f};

  {  // output irrep 0: 64x0e, ncols=1; paths P0,P4,P9
    v8f acc[4] = {vzero, vzero, vzero, vzero};
    path_mm<4, 64, 1>(wz + 0, Tw, 0, 64, lane, acc);
    path_mm<4, 64, 1>(wz + 8192, Tw, 224, 32, lane, acc);
    path_mm<4, 64, 1>(wz + 12544, Tw, 352, 16, lane, acc);
    if (nn == 0) {
#pragma unroll
      for (int t = 0; t < 4; ++t)
#pragma unroll
        for (int r = 0; r < 8; ++r) oz[t * 16 + mh + r] = acc[t][r];
    }
  }
  {  // output irrep 1: 32x1o, ncols=3; paths P1,P3,P6,P8
    v8f acc[2] = {vzero, vzero};
    path_mm<2, 32, 3>(wz + 4096, Tw, 64, 64, lane, acc);
    path_mm<2, 32, 3>(wz + 7168, Tw, 192, 32, lane, acc);
    path_mm<2, 32, 3>(wz + 10752, Tw, 288, 32, lane, acc);
    path_mm<2, 32, 3>(wz + 12032, Tw, 336, 16, lane, acc);
    if (nn < 3) {
#pragma unroll
      for (int t = 0; t < 2; ++t)
#pragma unroll
        for (int r = 0; r < 8; ++r)
          oz[64 + (t * 16 + mh + r) * 3 + nn] = acc[t][r];
    }
  }
  {  // output irrep 2: 16x2e, ncols=5; paths P2,P5,P7,P10
    v8f acc[1] = {vzero};
    path_mm<1, 16, 5>(wz + 6144, Tw, 128, 64, lane, acc);
    path_mm<1, 16, 5>(wz + 10240, Tw, 256, 32, lane, acc);
    path_mm<1, 16, 5>(wz + 11776, Tw, 320, 16, lane, acc);
    path_mm<1, 16, 5>(wz + 13568, Tw, 368, 16, lane, acc);
    if (nn < 5) {
#pragma unroll
      for (int r = 0; r < 8; ++r) oz[160 + (mh + r) * 5 + nn] = acc[0][r];
    }
  }
}

extern "C" void kernel_launch(void* const* d_in, const int* in_sizes, int n_in,
                              void* d_out, int out_size, void* d_ws,
                              size_t ws_size, hipStream_t stream) {
  (void)n_in; (void)out_size; (void)d_ws; (void)ws_size;
  const float* u = (const float*)d_in[0];
  const float* v = (const float*)d_in[1];
  const float* w = (const float*)d_in[2];
  float* out = (float*)d_out;
  const int nEdges = in_sizes[0] / 240;
  const int nBlocks = (nEdges + WAVES_PER_BLOCK - 1) / WAVES_PER_BLOCK;
  fctp_kernel<<<nBlocks, WAVES_PER_BLOCK * 32, 0, stream>>>(u, v, w, out,
                                                            nEdges);
}